// BodyStructureLoss_75625784148512
// MI455X (gfx1250) — compile-verified
//
#include <hip/hip_runtime.h>

typedef __attribute__((ext_vector_type(4))) float f4;
typedef __attribute__((ext_vector_type(2))) float v2f;
typedef __attribute__((ext_vector_type(8))) float v8f;

#define NB 2048
#define NT 256
#define HALF_BODY 1.0f

__device__ __forceinline__ float gated_norm(float x, float y, float z) {
    float d = sqrtf(fmaf(x, x, fmaf(y, y, z * z)));
    return d > HALF_BODY ? d : 0.0f;
}

// Pass 1: streaming gated-norm accumulation, 48B (4 joints) per thread-iter,
// 128-bit non-temporal loads, deterministic block reduction -> d_ws partials.
__global__ __launch_bounds__(NT) void bsl_partial_kernel(
    const float* __restrict__ in, float* __restrict__ partial,
    long long nchunk, long long njoint) {
    const int tid = threadIdx.x;
    const long long gtid = (long long)blockIdx.x * NT + tid;
    const long long stride = (long long)gridDim.x * NT;
    const f4* __restrict__ in4 = (const f4*)in;

    float acc = 0.0f;
    for (long long c = gtid; c < nchunk; c += stride) {
        const f4* p = in4 + 3 * c;                 // 48B aligned chunk: 4 joints
        f4 v0 = __builtin_nontemporal_load(p);
        f4 v1 = __builtin_nontemporal_load(p + 1);
        f4 v2 = __builtin_nontemporal_load(p + 2);
        acc += gated_norm(v0.x, v0.y, v0.z);
        acc += gated_norm(v0.w, v1.x, v1.y);
        acc += gated_norm(v1.z, v1.w, v2.x);
        acc += gated_norm(v2.y, v2.z, v2.w);
    }

    // Tail joints not covered by 4-joint chunks (none for the given shape).
    if (blockIdx.x == 0 && tid == 0) {
        for (long long j = nchunk * 4; j < njoint; ++j) {
            acc += gated_norm(in[3 * j], in[3 * j + 1], in[3 * j + 2]);
        }
    }

    // wave32 reduction (deterministic tree)
    #pragma unroll
    for (int off = 16; off > 0; off >>= 1)
        acc += __shfl_xor(acc, off, 32);

    __shared__ float lds[NT / 32];
    if ((tid & 31) == 0) lds[tid >> 5] = acc;
    __syncthreads();
    if (tid == 0) {
        float s = 0.0f;
        #pragma unroll
        for (int i = 0; i < NT / 32; ++i) s += lds[i];
        partial[blockIdx.x] = s;
    }
}

// Pass 2: single block reduces NB partials. Final 64->1 stage is done with
// V_WMMA_F32_16X16X4_F32: A = 64 residual partials (16x4 f32 A layout,
// 2 VGPRs per lane), B = ones(4x16)  =>  D[m][n] = rowsum(A)[m].
// Per the 16x16 f32 C/D layout, sum of a lane's 8 accumulator VGPRs gives
// sum(rowsum[0..7]) in lanes 0-15 and sum(rowsum[8..15]) in lanes 16-31,
// so total = lane0 + lane16. Exact (addition reordering only).
__global__ __launch_bounds__(NT) void bsl_final_kernel(
    const float* __restrict__ partial, float* __restrict__ out,
    long long njoint) {
    __shared__ float lds[NT];
    __shared__ float red[64];
    const int tid = threadIdx.x;

    float v = 0.0f;
    #pragma unroll
    for (int i = 0; i < NB / NT; ++i) v += partial[tid + i * NT];
    lds[tid] = v;
    __syncthreads();

    if (tid < 64)
        red[tid] = lds[tid] + lds[tid + 64] + lds[tid + 128] + lds[tid + 192];
    __syncthreads();

    if (tid < 32) {  // exactly wave 0: EXEC all ones, as WMMA requires
        v2f a;
        a.x = red[2 * tid];
        a.y = red[2 * tid + 1];
        v2f b;
        b.x = 1.0f;
        b.y = 1.0f;
        v8f c = {};
        c = __builtin_amdgcn_wmma_f32_16x16x4_f32(
            /*neg_a=*/false, a, /*neg_b=*/false, b,
            /*c_mod=*/(short)0, c, /*reuse_a=*/false, /*reuse_b=*/false);
        float s = c[0] + c[1] + c[2] + c[3] + c[4] + c[5] + c[6] + c[7];
        float total = __shfl(s, 0, 32) + __shfl(s, 16, 32);
        if (tid == 0) out[0] = total / (float)njoint;
    }
}

extern "C" void kernel_launch(void* const* d_in, const int* in_sizes, int n_in,
                              void* d_out, int out_size, void* d_ws, size_t ws_size,
                              hipStream_t stream) {
    const float* in = (const float*)d_in[0];
    float* out = (float*)d_out;
    float* partial = (float*)d_ws;  // NB floats = 8 KB scratch

    long long n = in_sizes[0];      // B*J*3 floats
    long long njoint = n / 3;       // B*J
    long long nchunk = njoint / 4;  // 4 joints (48B) per chunk

    bsl_partial_kernel<<<NB, NT, 0, stream>>>(in, partial, nchunk, njoint);
    bsl_final_kernel<<<1, NT, 0, stream>>>(partial, out, njoint);
}